// SAGE_35442070127065
// MI455X (gfx1250) — compile-verified
//
#include <hip/hip_runtime.h>

typedef __attribute__((ext_vector_type(2))) float v2f;
typedef __attribute__((ext_vector_type(8))) float v8f;

// ---------------- utility: zero fill ----------------
__global__ void zero_f32(float* __restrict__ p, long n) {
    long i = (long)blockIdx.x * blockDim.x + threadIdx.x;
    long stride = (long)gridDim.x * blockDim.x;
    for (; i < n; i += stride) p[i] = 0.0f;
}

// ---------------- per-edge degree count ----------------
__global__ void edge_count(const int* __restrict__ dst, int E, float* __restrict__ cnt) {
    int e = blockIdx.x * blockDim.x + threadIdx.x;
    if (e < E) atomicAdd(&cnt[dst[e]], 1.0f);
}

// ---------------- scatter-add: agg[dst] += x[src] ----------------
// d/4 lanes per edge, each lane moves one float4 (global_load_b128) and issues
// 4 contiguous f32 atomics. d=128 -> 1 edge/wave, d=64 -> 2 edges/wave.
__global__ void edge_scatter(const float* __restrict__ x,
                             const int* __restrict__ src,
                             const int* __restrict__ dst,
                             int E, int d, float* __restrict__ agg) {
    int lpe  = d >> 2;                     // lanes per edge
    int lane = threadIdx.x & 31;
    int wave = blockIdx.x * (blockDim.x >> 5) + (threadIdx.x >> 5);
    int epw  = 32 / lpe;                   // edges per wave
    long e   = (long)wave * epw + lane / lpe;
    int  fl  = lane % lpe;                 // float4 slot within the row
    if (e >= E) return;
    int s = src[e];
    int t = dst[e];
    const float4* xs = (const float4*)(x + (long)s * d);
    float4 v = xs[fl];
    float* ag = agg + (long)t * d + fl * 4;
    atomicAdd(ag + 0, v.x);
    atomicAdd(ag + 1, v.y);
    atomicAdd(ag + 2, v.z);
    atomicAdd(ag + 3, v.w);
}

// ---------------- agg[row][*] /= max(cnt[row], 1) ----------------
__global__ void row_mean(float* __restrict__ agg, const float* __restrict__ cnt, int d) {
    int row = blockIdx.x;
    float inv = 1.0f / fmaxf(cnt[row], 1.0f);
    agg[(long)row * d + threadIdx.x] *= inv;
}

// ---------------- fused SAGE GEMM: Out = Am@Bl + Ax@Br + bias (opt ReLU) ----------------
// One wave computes one 16x16 output tile via chained V_WMMA_F32_16X16X4_F32.
// Both weight matrices are staged in LDS once per workgroup (K*M == 8192 for
// both layers). blockDim.x must be (M/16)*32; grid.x = ceil(Nn/16).
__global__ void sage_wmma_gemm(const float* __restrict__ Am,   // [Nn, K] mean-aggregated
                               const float* __restrict__ Ax,   // [Nn, K] self features
                               const float* __restrict__ Bl,   // [K, M]
                               const float* __restrict__ Br,   // [K, M]
                               const float* __restrict__ bias, // [M]
                               float* __restrict__ Out,        // [Nn, M]
                               int Nn, int K, int M, int relu) {
    __shared__ float sBl[8192];
    __shared__ float sBr[8192];
    const int KM = K * M;
    for (int i = threadIdx.x; i < KM; i += blockDim.x) {
        sBl[i] = Bl[i];
        sBr[i] = Br[i];
    }
    __syncthreads();

    const int lane = threadIdx.x & 31;
    const int h    = lane >> 4;   // lane half (0: lanes 0-15, 1: lanes 16-31)
    const int l    = lane & 15;
    const int colb = (threadIdx.x >> 5) * 16;
    const int col  = colb + l;

    int row  = blockIdx.x * 16 + l;            // A-matrix row for this lane
    int rowc = row < Nn ? row : Nn - 1;        // clamp so EXEC stays all-ones

    // C/D layout: lane (h,l), VGPR v -> D[row = v + 8h][col = l]; bias depends on col only.
    v8f acc;
    float bv = bias[col];
#pragma unroll
    for (int v = 0; v < 8; ++v) acc[v] = bv;

    const float* am = Am + (long)rowc * K;
    const float* ax = Ax + (long)rowc * K;

    for (int k0 = 0; k0 < K; k0 += 4) {
        int ka = k0 + 2 * h;  // this lane covers K = ka, ka+1
        // term 1: mean @ Bl
        v2f a1; a1.x = am[ka]; a1.y = am[ka + 1];
        v2f b1; b1.x = sBl[ka * M + col]; b1.y = sBl[(ka + 1) * M + col];
        acc = __builtin_amdgcn_wmma_f32_16x16x4_f32(false, a1, false, b1,
                                                    (short)0, acc, false, false);
        // term 2: x @ Br
        v2f a2; a2.x = ax[ka]; a2.y = ax[ka + 1];
        v2f b2; b2.x = sBr[ka * M + col]; b2.y = sBr[(ka + 1) * M + col];
        acc = __builtin_amdgcn_wmma_f32_16x16x4_f32(false, a2, false, b2,
                                                    (short)0, acc, false, false);
    }

    int rb = blockIdx.x * 16;
#pragma unroll
    for (int v = 0; v < 8; ++v) {
        int r = rb + v + 8 * h;
        if (r < Nn) {
            float val = acc[v];
            if (relu) val = fmaxf(val, 0.0f);
            Out[(long)r * M + col] = val;
        }
    }
}

extern "C" void kernel_launch(void* const* d_in, const int* in_sizes, int n_in,
                              void* d_out, int out_size, void* d_ws, size_t ws_size,
                              hipStream_t stream) {
    const float* x   = (const float*)d_in[0];
    const int*   ei  = (const int*)d_in[1];
    const float* w1l = (const float*)d_in[2];
    const float* b1  = (const float*)d_in[3];
    const float* w1r = (const float*)d_in[4];
    const float* w2l = (const float*)d_in[5];
    const float* b2  = (const float*)d_in[6];
    const float* w2r = (const float*)d_in[7];
    float* out = (float*)d_out;

    const int IN = 128, HID = 64;
    long Nn  = (long)in_sizes[0] / IN;     // 100000
    int  E   = in_sizes[1] / 2;            // 1600000
    int  OUT = in_sizes[5] / HID;          // 128
    const int* src = ei;
    const int* dst = ei + E;

    // workspace layout: cnt[Nn] (padded) | agg[Nn*IN] | h[Nn*HID]
    long cntPad = (Nn + 511) & ~511L;
    float* cnt  = (float*)d_ws;
    float* agg  = cnt + cntPad;
    float* hbuf = agg + Nn * IN;

    // scatter launch geometry: edges-per-wave depends on feature width
    int epw1 = 32 / (IN >> 2);   // 1 edge per wave  (d=128)
    int epw2 = 32 / (HID >> 2);  // 2 edges per wave (d=64)
    int blk1 = (E + epw1 * 8 - 1) / (epw1 * 8);
    int blk2 = (E + epw2 * 8 - 1) / (epw2 * 8);

    // ---- layer 1 ----
    zero_f32<<<4096, 256, 0, stream>>>((float*)d_ws, cntPad + Nn * IN);
    edge_count<<<(E + 255) / 256, 256, 0, stream>>>(dst, E, cnt);
    edge_scatter<<<blk1, 256, 0, stream>>>(x, src, dst, E, IN, agg);
    row_mean<<<(int)Nn, IN, 0, stream>>>(agg, cnt, IN);
    sage_wmma_gemm<<<(int)((Nn + 15) / 16), (HID / 16) * 32, 0, stream>>>(
        agg, x, w1l, w1r, b1, hbuf, (int)Nn, IN, HID, 1);

    // ---- layer 2 ----
    zero_f32<<<4096, 256, 0, stream>>>(agg, Nn * HID);
    edge_scatter<<<blk2, 256, 0, stream>>>(hbuf, src, dst, E, HID, agg);
    row_mean<<<(int)Nn, HID, 0, stream>>>(agg, cnt, HID);
    sage_wmma_gemm<<<(int)((Nn + 15) / 16), (OUT / 16) * 32, 0, stream>>>(
        agg, hbuf, w2l, w2r, b2, out, (int)Nn, HID, OUT, 0);
}